// SelfAttention_13881334301013
// MI455X (gfx1250) — compile-verified
//
#include <hip/hip_runtime.h>
#include <hip/hip_bf16.h>
#include <math.h>

// ---------------------------------------------------------------------------
// Self-attention w/ banded mask, entropy + diversity side outputs.
// MI455X (gfx1250) plan:
//  - All GEMMs on V_WMMA_F32_16X16X4_F32 (fp32-exact tensor path).
//  - Intermediates (K,Q,V = 48MB, logits = 64MB) live in workspace; the 192MB
//    L2 keeps them resident, so two row-wise re-reads of logits are ~free.
//  - att^T @ V restricted to the |i-j|<250 band (8x fewer FLOPs).
//  - Outputs (~220MB) are the HBM floor (~9.5us @ 23.3 TB/s).
// ---------------------------------------------------------------------------

typedef __attribute__((ext_vector_type(2))) float v2f;
typedef __attribute__((ext_vector_type(8))) float v8f;

#define TDIM 4096
#define MDIM 1024
#define DDIM 1024
#define APP  250      // |i-j| >= APP masked
#define QSCALE 0.06f

// ---------------- reductions (wave32) ----------------
__device__ __forceinline__ float wave_max(float v) {
  for (int o = 16; o > 0; o >>= 1) v = fmaxf(v, __shfl_xor(v, o, 32));
  return v;
}
__device__ __forceinline__ float wave_sum(float v) {
  for (int o = 16; o > 0; o >>= 1) v += __shfl_xor(v, o, 32);
  return v;
}
__device__ __forceinline__ float wave_prod(float v) {
  for (int o = 16; o > 0; o >>= 1) v *= __shfl_xor(v, o, 32);
  return v;
}
// mode: 0 = max, 1 = sum, 2 = prod. All threads return the result.
__device__ __forceinline__ float block_reduce(float v, float* sm, int mode) {
  v = (mode == 0) ? wave_max(v) : (mode == 1) ? wave_sum(v) : wave_prod(v);
  int w = threadIdx.x >> 5, l = threadIdx.x & 31;
  __syncthreads();
  if (l == 0) sm[w] = v;
  __syncthreads();
  int nw = blockDim.x >> 5;
  float r = (mode == 0) ? -INFINITY : (mode == 1) ? 0.0f : 1.0f;
  for (int i = 0; i < nw; ++i) {
    float u = sm[i];
    r = (mode == 0) ? fmaxf(r, u) : (mode == 1) ? (r + u) : (r * u);
  }
  return r;
}

// ---------------- WMMA helper ----------------
__device__ __forceinline__ v8f wmma4(v2f a, v2f b, v8f c) {
  // D = A(16x4 f32) * B(4x16 f32) + C(16x16 f32)
  return __builtin_amdgcn_wmma_f32_16x16x4_f32(
      /*neg_a=*/false, a, /*neg_b=*/false, b,
      /*c_mod=*/(short)0, c, /*reuse_a=*/false, /*reuse_b=*/false);
}

// ---------------------------------------------------------------------------
// K1: K = x@Wk ; Q = 0.06*(x@Wq) ; V = x@Wv.  blockIdx.z selects which.
// Per-wave tile 32 rows x 64 cols; block = 4 waves -> 128 rows x 64 cols.
// grid = (DDIM/64, TDIM/128, 3), block = 128 threads.
// ---------------------------------------------------------------------------
__global__ void proj_kernel(const float* __restrict__ x,
                            const float* __restrict__ Wk,
                            const float* __restrict__ Wq,
                            const float* __restrict__ Wv,
                            float* __restrict__ Ko, float* __restrict__ Qo,
                            float* __restrict__ Vo) {
  const float* W; float* C; float scale;
  if (blockIdx.z == 0)      { W = Wk; C = Ko; scale = 1.0f;   }
  else if (blockIdx.z == 1) { W = Wq; C = Qo; scale = QSCALE; }
  else                      { W = Wv; C = Vo; scale = 1.0f;   }

  const int lane = threadIdx.x & 31;
  const int wv   = threadIdx.x >> 5;
  const int half = lane & 15;
  const int koff = (lane >> 4) << 1;   // lanes 16..31 hold K+2,K+3
  const int rowBase = blockIdx.y * 128 + wv * 32;
  const int colBase = blockIdx.x * 64;

  v8f acc[2][4] = {};
  for (int k = 0; k < MDIM; k += 4) {
    const float* xp = x + (size_t)(rowBase + half) * MDIM + k + koff;
    v2f a[2];
    a[0] = *(const v2f*)xp;
    a[1] = *(const v2f*)(xp + 16 * MDIM);
    v2f b[4];
#pragma unroll
    for (int ct = 0; ct < 4; ++ct) {
      const float* wp = W + (size_t)(k + koff) * DDIM + colBase + 16 * ct + half;
      b[ct].x = wp[0];
      b[ct].y = wp[DDIM];
    }
#pragma unroll
    for (int rt = 0; rt < 2; ++rt)
#pragma unroll
      for (int ct = 0; ct < 4; ++ct)
        acc[rt][ct] = wmma4(a[rt], b[ct], acc[rt][ct]);
  }

  const int rAdd = (lane >> 4) << 3;   // lanes 16..31 -> rows +8
#pragma unroll
  for (int rt = 0; rt < 2; ++rt)
#pragma unroll
    for (int r = 0; r < 8; ++r) {
      int row = rowBase + 16 * rt + r + rAdd;
      float* cp = C + (size_t)row * DDIM + colBase + half;
#pragma unroll
      for (int ct = 0; ct < 4; ++ct) cp[16 * ct] = acc[rt][ct][r] * scale;
    }
}

// ---------------------------------------------------------------------------
// K2: logits = Q @ K^T  (TDIM x TDIM, contraction over DDIM).
// B[k][n] = K[n][k] -> contiguous float2 loads from K rows.
// grid = (TDIM/64, TDIM/128), block = 128.
// ---------------------------------------------------------------------------
__global__ void logits_kernel(const float* __restrict__ Q,
                              const float* __restrict__ Km,
                              float* __restrict__ Lg) {
  const int lane = threadIdx.x & 31;
  const int wv   = threadIdx.x >> 5;
  const int half = lane & 15;
  const int koff = (lane >> 4) << 1;
  const int rowBase = blockIdx.y * 128 + wv * 32;
  const int colBase = blockIdx.x * 64;

  v8f acc[2][4] = {};
  for (int k = 0; k < DDIM; k += 4) {
    const float* qp = Q + (size_t)(rowBase + half) * DDIM + k + koff;
    v2f a[2];
    a[0] = *(const v2f*)qp;
    a[1] = *(const v2f*)(qp + 16 * DDIM);
    v2f b[4];
#pragma unroll
    for (int ct = 0; ct < 4; ++ct)
      b[ct] = *(const v2f*)(Km + (size_t)(colBase + 16 * ct + half) * DDIM + k + koff);
#pragma unroll
    for (int rt = 0; rt < 2; ++rt)
#pragma unroll
      for (int ct = 0; ct < 4; ++ct)
        acc[rt][ct] = wmma4(a[rt], b[ct], acc[rt][ct]);
  }

  const int rAdd = (lane >> 4) << 3;
#pragma unroll
  for (int rt = 0; rt < 2; ++rt)
#pragma unroll
    for (int r = 0; r < 8; ++r) {
      int row = rowBase + 16 * rt + r + rAdd;
      float* cp = Lg + (size_t)row * TDIM + colBase + half;
#pragma unroll
      for (int ct = 0; ct < 4; ++ct) cp[16 * ct] = acc[rt][ct][r];
    }
}

// ---------------------------------------------------------------------------
// K3: entropy over UNMASKED logits rows. entropy = (log s1 - s2/s1)/log(T)
// with s1 = sum exp(l-m), s2 = sum (l-m)exp(l-m). One block per row.
// ---------------------------------------------------------------------------
__global__ void entropy_kernel(const float* __restrict__ Lg,
                               float* __restrict__ ent) {
  __shared__ float sm[8];
  const int i = blockIdx.x;
  const float* row = Lg + (size_t)i * TDIM;
  float m = -INFINITY;
  for (int j = threadIdx.x; j < TDIM; j += blockDim.x) m = fmaxf(m, row[j]);
  m = block_reduce(m, sm, 0);
  float s1 = 0.0f, s2 = 0.0f;
  for (int j = threadIdx.x; j < TDIM; j += blockDim.x) {
    float l = row[j] - m;
    float e = expf(l);
    s1 += e;
    s2 += l * e;
  }
  s1 = block_reduce(s1, sm, 1);
  s2 = block_reduce(s2, sm, 1);
  if (threadIdx.x == 0)
    ent[i] = (logf(s1) - s2 / s1) * (1.0f / logf((float)TDIM));
}

// ---------------------------------------------------------------------------
// K4: banded masked softmax -> att row (zeros outside band), and
// divh[i] = prod over band of (1 - att). One block per row.
// ---------------------------------------------------------------------------
__global__ void att_kernel(const float* __restrict__ Lg,
                           float* __restrict__ att,
                           float* __restrict__ divh) {
  __shared__ float sm[8];
  const int i = blockIdx.x;
  const float* row = Lg + (size_t)i * TDIM;
  float* arow = att + (size_t)i * TDIM;
  int jlo = i - (APP - 1); if (jlo < 0) jlo = 0;
  int jhi = i + (APP - 1); if (jhi > TDIM - 1) jhi = TDIM - 1;

  for (int j = threadIdx.x; j < TDIM; j += blockDim.x) arow[j] = 0.0f;
  __syncthreads();

  float m = -INFINITY;
  for (int j = jlo + threadIdx.x; j <= jhi; j += blockDim.x)
    if (j != i) m = fmaxf(m, row[j]);
  m = block_reduce(m, sm, 0);

  float s = 0.0f;
  for (int j = jlo + threadIdx.x; j <= jhi; j += blockDim.x)
    if (j != i) s += expf(row[j] - m);
  s = block_reduce(s, sm, 1);
  const float inv = 1.0f / s;

  float p = 1.0f;
  for (int j = jlo + threadIdx.x; j <= jhi; j += blockDim.x)
    if (j != i) {
      float a = expf(row[j] - m) * inv;
      arow[j] = a;
      p *= (1.0f - a);
    }
  p = block_reduce(p, sm, 2);
  if (threadIdx.x == 0) divh[i] = p;
}

// ---------------------------------------------------------------------------
// K5: div = divh / sum(|divh|). Single block.
// ---------------------------------------------------------------------------
__global__ void divnorm_kernel(const float* __restrict__ divh,
                               float* __restrict__ divv) {
  __shared__ float sm[8];
  float s = 0.0f;
  for (int j = threadIdx.x; j < TDIM; j += blockDim.x) s += fabsf(divh[j]);
  s = block_reduce(s, sm, 1);
  const float inv = 1.0f / s;
  for (int j = threadIdx.x; j < TDIM; j += blockDim.x) divv[j] = divh[j] * inv;
}

// ---------------------------------------------------------------------------
// K6: broadcast windows: ent_win[i][j] = ent[j], div_win[i][j] = div[j].
// float4 streaming writes (pure HBM bandwidth).
// ---------------------------------------------------------------------------
__global__ void bcast_kernel(const float* __restrict__ ent,
                             const float* __restrict__ divv,
                             float* __restrict__ entw,
                             float* __restrict__ divw) {
  size_t idx = (size_t)blockIdx.x * blockDim.x + threadIdx.x;
  const size_t n4 = (size_t)TDIM * TDIM / 4;
  if (idx >= n4) return;
  int j4 = (int)(idx & (TDIM / 4 - 1));
  ((float4*)entw)[idx] = ((const float4*)ent)[j4];
  ((float4*)divw)[idx] = ((const float4*)divv)[j4];
}

// ---------------------------------------------------------------------------
// K7: Z = att^T @ V  (banded: only j with |i-j| <= 249 contribute).
// A[m][k] = att[k][m] (column loads); grid = (DDIM/64, TDIM/128).
// ---------------------------------------------------------------------------
__global__ void attv_kernel(const float* __restrict__ att,
                            const float* __restrict__ V,
                            float* __restrict__ Z) {
  const int lane = threadIdx.x & 31;
  const int wv   = threadIdx.x >> 5;
  const int half = lane & 15;
  const int koff = (lane >> 4) << 1;
  const int rowBase = blockIdx.y * 128 + wv * 32;
  const int colBase = blockIdx.x * 64;

  int klo = rowBase - 252; if (klo < 0) klo = 0;          // mult of 4
  int khi = rowBase + 32 + 252; if (khi > TDIM) khi = TDIM;

  v8f acc[2][4] = {};
  const int m0 = rowBase + half;
  for (int k = klo; k < khi; k += 4) {
    v2f a[2];
    a[0].x = att[(size_t)(k + koff) * TDIM + m0];
    a[0].y = att[(size_t)(k + koff + 1) * TDIM + m0];
    a[1].x = att[(size_t)(k + koff) * TDIM + m0 + 16];
    a[1].y = att[(size_t)(k + koff + 1) * TDIM + m0 + 16];
    v2f b[4];
#pragma unroll
    for (int ct = 0; ct < 4; ++ct) {
      const float* vp = V + (size_t)(k + koff) * DDIM + colBase + 16 * ct + half;
      b[ct].x = vp[0];
      b[ct].y = vp[DDIM];
    }
#pragma unroll
    for (int rt = 0; rt < 2; ++rt)
#pragma unroll
      for (int ct = 0; ct < 4; ++ct)
        acc[rt][ct] = wmma4(a[rt], b[ct], acc[rt][ct]);
  }

  const int rAdd = (lane >> 4) << 3;
#pragma unroll
  for (int rt = 0; rt < 2; ++rt)
#pragma unroll
    for (int r = 0; r < 8; ++r) {
      int row = rowBase + 16 * rt + r + rAdd;
      float* cp = Z + (size_t)row * DDIM + colBase + half;
#pragma unroll
      for (int ct = 0; ct < 4; ++ct) cp[16 * ct] = acc[rt][ct][r];
    }
}

// ---------------------------------------------------------------------------
// K8: y = Z @ Wout  (TDIM x MDIM, contraction over DDIM). grid=(MDIM/64,TDIM/128)
// ---------------------------------------------------------------------------
__global__ void outgemm_kernel(const float* __restrict__ Z,
                               const float* __restrict__ Wout,
                               float* __restrict__ y) {
  const int lane = threadIdx.x & 31;
  const int wv   = threadIdx.x >> 5;
  const int half = lane & 15;
  const int koff = (lane >> 4) << 1;
  const int rowBase = blockIdx.y * 128 + wv * 32;
  const int colBase = blockIdx.x * 64;

  v8f acc[2][4] = {};
  for (int k = 0; k < DDIM; k += 4) {
    const float* zp = Z + (size_t)(rowBase + half) * DDIM + k + koff;
    v2f a[2];
    a[0] = *(const v2f*)zp;
    a[1] = *(const v2f*)(zp + 16 * DDIM);
    v2f b[4];
#pragma unroll
    for (int ct = 0; ct < 4; ++ct) {
      const float* wp = Wout + (size_t)(k + koff) * MDIM + colBase + 16 * ct + half;
      b[ct].x = wp[0];
      b[ct].y = wp[MDIM];
    }
#pragma unroll
    for (int rt = 0; rt < 2; ++rt)
#pragma unroll
      for (int ct = 0; ct < 4; ++ct)
        acc[rt][ct] = wmma4(a[rt], b[ct], acc[rt][ct]);
  }

  const int rAdd = (lane >> 4) << 3;
#pragma unroll
  for (int rt = 0; rt < 2; ++rt)
#pragma unroll
    for (int r = 0; r < 8; ++r) {
      int row = rowBase + 16 * rt + r + rAdd;
      float* cp = y + (size_t)row * MDIM + colBase + half;
#pragma unroll
      for (int ct = 0; ct < 4; ++ct) cp[16 * ct] = acc[rt][ct][r];
    }
}

// ---------------------------------------------------------------------------
extern "C" void kernel_launch(void* const* d_in, const int* in_sizes, int n_in,
                              void* d_out, int out_size, void* d_ws, size_t ws_size,
                              hipStream_t stream) {
  const float* x    = (const float*)d_in[0];
  const float* Wk   = (const float*)d_in[1];
  const float* Wq   = (const float*)d_in[2];
  const float* Wv   = (const float*)d_in[3];
  const float* Wout = (const float*)d_in[4];

  float* out = (float*)d_out;
  const size_t TD = (size_t)TDIM * DDIM;   // 4M
  const size_t TT = (size_t)TDIM * TDIM;   // 16M
  float* y_out   = out;
  float* att_out = out + (size_t)TDIM * MDIM;
  float* ent_win = att_out + TT;
  float* div_win = ent_win + TT;

  float* ws   = (float*)d_ws;
  float* Kp   = ws;             // [T,D]
  float* Qp   = ws + TD;        // [T,D]
  float* Vp   = ws + 2 * TD;    // [T,D]
  float* Lg   = ws + 3 * TD;    // [T,T] logits
  float* ent  = Lg + TT;        // [T]
  float* divh = ent + TDIM;     // [T]
  float* divv = divh + TDIM;    // [T]
  float* Zp   = Kp;             // reuse K slot after logits done: [T,D]

  dim3 blk(128);

  // K,Q,V projections
  proj_kernel<<<dim3(DDIM / 64, TDIM / 128, 3), blk, 0, stream>>>(
      x, Wk, Wq, Wv, Kp, Qp, Vp);

  // logits = Q @ K^T
  logits_kernel<<<dim3(TDIM / 64, TDIM / 128), blk, 0, stream>>>(Qp, Kp, Lg);

  // row-wise entropy of unmasked logits
  entropy_kernel<<<TDIM, 256, 0, stream>>>(Lg, ent);

  // banded masked softmax -> att, and row products -> divh
  att_kernel<<<TDIM, 256, 0, stream>>>(Lg, att_out, divh);

  // div normalize
  divnorm_kernel<<<1, 256, 0, stream>>>(divh, divv);

  // broadcast windows
  {
    size_t n4 = TT / 4;
    int nb = (int)((n4 + 255) / 256);
    bcast_kernel<<<nb, 256, 0, stream>>>(ent, divv, ent_win, div_win);
  }

  // Z = att^T @ V (banded), then y = Z @ Wout
  attv_kernel<<<dim3(DDIM / 64, TDIM / 128), blk, 0, stream>>>(att_out, Vp, Zp);
  outgemm_kernel<<<dim3(MDIM / 64, TDIM / 128), blk, 0, stream>>>(Zp, Wout, y_out);
}